// IWSCL_86285892976910
// MI455X (gfx1250) — compile-verified
//
#include <hip/hip_runtime.h>
#include <hip/hip_bf16.h>
#include <math.h>

// ---------------------------------------------------------------------------
// IWSCL fused loss on MI455X (gfx1250, wave32, WMMA + async-to-LDS).
// Memory-bound: queue (128MB) read exactly once from HBM via async LDS DMA;
// features (2MB) stay L2-resident and are double-buffered per b-tile.
// FP32 WMMA 16x16x4 keeps reference precision (no reason to down-convert:
// 68.7 GFLOP vs 128MB/23.3TB/s ~ 5.5us -> bandwidth bound).
// ---------------------------------------------------------------------------

typedef __attribute__((ext_vector_type(2))) float v2f;
typedef __attribute__((ext_vector_type(8))) float v8f;

#define B_TOT   1024
#define Q_TOT   65536
#define D_TOT   512
#define INV_T   (1.0f / 0.07f)

#define QT      64          // queue rows per block
#define BT      32          // feature rows per inner iteration
#define STRIDE  516         // padded floats per LDS row (516 % 64 == 4 -> conflict-free b64)
#define NTHR    256         // 8 waves

// Async DMA: global -> LDS, 16B per lane, tracked by ASYNCcnt (bypasses VGPRs).
// GV addressing: 64-bit address in VGPR pair, saddr = off.
__device__ __forceinline__ void async_b128(unsigned lds_off, const float* g) {
    asm volatile("global_load_async_to_lds_b128 %0, %1, off"
                 :: "v"(lds_off), "v"((unsigned long long)(uintptr_t)g)
                 : "memory");
}
// Async loads retire in order: waiting ASYNCcnt<=16 means everything older
// than the 16 most recently issued ops (the in-flight prefetch) has landed.
#define WAIT_ASYNC_0()  asm volatile("s_wait_asynccnt 0x0"  ::: "memory")
#define WAIT_ASYNC_16() asm volatile("s_wait_asynccnt 0x10" ::: "memory")

// ---------------------------------------------------------------------------
// prep: zero accumulators, pred labels, zero label histogram
__global__ void iwscl_prep(const float* __restrict__ ipred,
                           const int*   __restrict__ ilab,
                           int* __restrict__ pred,
                           float* __restrict__ pos,
                           float* __restrict__ neg,
                           int* __restrict__ cnt) {
    int t = threadIdx.x;                 // 1024 threads
    pos[t] = 0.0f;
    neg[t] = 0.0f;
    if (t < 2) cnt[t] = 0;
    int l = ilab[t];
    pred[t] = (l != -1) ? l : (ipred[t] > 0.5f ? 1 : 0);
}

// histogram of queue labels (for family-count validity test)
__global__ void iwscl_count(const int* __restrict__ qlab, int* __restrict__ cnt) {
    int i = blockIdx.x * blockDim.x + threadIdx.x;
    atomicAdd(&cnt[qlab[i]], 1);
}

// ---------------------------------------------------------------------------
// Main fused kernel: normalize(queue) . features^T, exp, masked row sums.
// One block owns QT queue rows for the whole B dimension.
__global__ __launch_bounds__(NTHR) void iwscl_gemm(
        const float* __restrict__ feat,     // [B,512]
        const float* __restrict__ queue,    // [Q,512]
        const int*   __restrict__ qlab,     // [Q]
        const int*   __restrict__ pred,     // [B]
        float* __restrict__ pos,            // [B]
        float* __restrict__ neg) {          // [B]
    extern __shared__ float lds[];
    float* lds_q   = lds;                        // QT*STRIDE
    float* lds_f0  = lds_q  + QT * STRIDE;       // BT*STRIDE (ping)
    float* lds_f1  = lds_f0 + BT * STRIDE;       // BT*STRIDE (pong)
    float* lds_inv = lds_f1 + BT * STRIDE;       // QT
    float* lds_pos = lds_inv + QT;               // BT
    float* lds_neg = lds_pos + BT;               // BT

    const int tid = threadIdx.x;
    const int q0  = blockIdx.x * QT;

    // ---- async-stage queue tile: QT*512 floats = 8192 x b128, 32 per thread
    #pragma unroll
    for (int i = 0; i < (QT * D_TOT) / (4 * NTHR); ++i) {
        int idx = tid + i * NTHR;                // b128 index
        int row = idx >> 7;                      // / (512/4)
        int c4  = idx & 127;
        async_b128((unsigned)(uintptr_t)&lds_q[row * STRIDE + c4 * 4],
                   queue + (size_t)(q0 + row) * D_TOT + c4 * 4);
    }
    // ---- async-stage feature tile 0 into ping buffer (16 per thread)
    #pragma unroll
    for (int i = 0; i < (BT * D_TOT) / (4 * NTHR); ++i) {
        int idx = tid + i * NTHR;
        int row = idx >> 7;
        int c4  = idx & 127;
        async_b128((unsigned)(uintptr_t)&lds_f0[row * STRIDE + c4 * 4],
                   feat + (size_t)row * D_TOT + c4 * 4);
    }
    WAIT_ASYNC_0();
    __syncthreads();

    // ---- per-row inverse L2 norm (4 threads per row) ----
    {
        int row = tid >> 2, part = tid & 3;
        const float* rp = &lds_q[row * STRIDE + part * 128];
        float s = 0.0f;
        #pragma unroll 8
        for (int j = 0; j < 128; ++j) s += rp[j] * rp[j];
        s += __shfl_xor(s, 1);
        s += __shfl_xor(s, 2);
        if (part == 0) lds_inv[row] = 1.0f / fmaxf(sqrtf(s), 1e-12f);
    }

    const int lane  = tid & 31;
    const int wave  = tid >> 5;          // 0..7
    const int qw    = wave & 3;          // which 16-wide Q subtile
    const int bw    = wave >> 2;         // which 16-wide B subtile (0..1)
    const int mrow  = lane & 15;
    const int khalf = (lane >> 4) ? 2 : 0;   // lanes 16..31 carry K+2,K+3

    for (int b0 = 0; b0 < B_TOT; b0 += BT) {
        float* cur = ((b0 >> 5) & 1) ? lds_f1 : lds_f0;
        float* nxt = ((b0 >> 5) & 1) ? lds_f0 : lds_f1;

        // Fence: prev iteration's WMMA reads of `nxt` (its cur) are done,
        // and prev lds_pos global-adds have read their values.
        __syncthreads();
        if (tid < BT) { lds_pos[tid] = 0.0f; lds_neg[tid] = 0.0f; }

        // Prefetch next feature tile into the alternate buffer, THEN wait for
        // the current tile (oldest 16 async ops) -> compute/fetch overlap.
        const int b0n = b0 + BT;
        if (b0n < B_TOT) {
            #pragma unroll
            for (int i = 0; i < (BT * D_TOT) / (4 * NTHR); ++i) {
                int idx = tid + i * NTHR;
                int row = idx >> 7;
                int c4  = idx & 127;
                async_b128((unsigned)(uintptr_t)&nxt[row * STRIDE + c4 * 4],
                           feat + (size_t)(b0n + row) * D_TOT + c4 * 4);
            }
            WAIT_ASYNC_16();
        } else {
            WAIT_ASYNC_0();
        }
        __syncthreads();   // all waves' `cur` tile + zeroed lds_pos visible

        // ---- 16x16 tile, K=512 via 128x V_WMMA_F32_16X16X4_F32 ----
        // A frag (16x4 f32): lane<16 -> {A[m][k],A[m][k+1]}, lane>=16 -> {A[m][k+2],A[m][k+3]}
        // B frag assumed mirror layout (row n of queue = column n of B).
        const float* aptr = &cur[(bw * 16 + mrow) * STRIDE + khalf];
        const float* bptr = &lds_q[(qw * 16 + mrow) * STRIDE + khalf];
        v8f acc0 = {};
        v8f acc1 = {};
        #pragma unroll 4
        for (int k = 0; k < D_TOT; k += 8) {
            v2f a0 = *reinterpret_cast<const v2f*>(aptr + k);
            v2f b0v = *reinterpret_cast<const v2f*>(bptr + k);
            acc0 = __builtin_amdgcn_wmma_f32_16x16x4_f32(
                false, a0, false, b0v, (short)0, acc0, false, false);
            v2f a1 = *reinterpret_cast<const v2f*>(aptr + k + 4);
            v2f b1v = *reinterpret_cast<const v2f*>(bptr + k + 4);
            acc1 = __builtin_amdgcn_wmma_f32_16x16x4_f32(
                false, a1, false, b1v, (short)0, acc1, false, false);
        }

        // ---- epilogue: exp + family-masked sums ----
        // C layout: lane = column n (mod 16), VGPR r = row r (lanes 0-15) / r+8 (lanes 16-31)
        int   q     = q0 + qw * 16 + mrow;
        float scale = lds_inv[qw * 16 + mrow] * INV_T;
        int   ql    = qlab[q];
        int   mbase = (lane >> 4) * 8;
        #pragma unroll
        for (int r = 0; r < 8; ++r) {
            int   m = r + mbase;
            int   b = b0 + bw * 16 + m;
            float e = expf((acc0[r] + acc1[r]) * scale);
            int   fam = (ql == pred[b]);
            float p  = fam ? e : 0.0f;
            float ng = fam ? 0.0f : e;
            // reduce over the 16 lanes holding this row (masks<16 keep halves apart)
            #pragma unroll
            for (int s = 1; s < 16; s <<= 1) {
                p  += __shfl_xor(p,  s);
                ng += __shfl_xor(ng, s);
            }
            if (mrow == 0) {
                atomicAdd(&lds_pos[bw * 16 + m], p);   // ds_add_f32
                atomicAdd(&lds_neg[bw * 16 + m], ng);
            }
        }
        __syncthreads();
        if (tid < BT) {
            atomicAdd(&pos[b0 + tid], lds_pos[tid]);   // global_atomic_add_f32
            atomicAdd(&neg[b0 + tid], lds_neg[tid]);
        }
    }
}

// ---------------------------------------------------------------------------
// finalize: loss reduction + prototype pseudo-labels (C=2 -> trivial GEMM)
__global__ void iwscl_final(const float* __restrict__ pos,
                            const float* __restrict__ neg,
                            const int*   __restrict__ cnt,
                            const int*   __restrict__ pred,
                            const int*   __restrict__ ilab,
                            const float* __restrict__ feat,
                            const float* __restrict__ protos,  // [2,512]
                            const float* __restrict__ pcc,     // [2,2]
                            float* __restrict__ out) {
    __shared__ float sl[B_TOT];
    __shared__ int   sv[B_TOT];
    int b = threadIdx.x;

    float p = pos[b], ng = neg[b];
    int   fc    = cnt[pred[b]];
    int   valid = (fc > 0) && (fc < Q_TOT);
    float ratio = p / (p + ng + 1e-6f);
    sl[b] = valid ? -logf(ratio) : 0.0f;
    sv[b] = valid;
    __syncthreads();
    #pragma unroll
    for (int s = 512; s > 0; s >>= 1) {
        if (b < s) { sl[b] += sl[b + s]; sv[b] += sv[b + s]; }
        __syncthreads();
    }
    if (b == 0) {
        int nv = sv[0] > 1 ? sv[0] : 1;
        out[0] = sl[0] / (float)nv;
    }

    // predict(): nearest prototype -> class
    const float* f = feat + (size_t)b * D_TOT;
    float s0 = 0.0f, s1 = 0.0f;
    #pragma unroll 8
    for (int j = 0; j < D_TOT; ++j) {
        float v = f[j];
        s0 += v * protos[j];
        s1 += v * protos[D_TOT + j];
    }
    int pc0 = (pcc[1] > pcc[0]) ? 1 : 0;       // argmax row 0 (first-index tie rule)
    int pc1 = (pcc[3] > pcc[2]) ? 1 : 0;       // argmax row 1
    int pl  = (s1 > s0) ? pc1 : pc0;
    int l   = ilab[b];
    out[1 + b] = (float)((l != -1) ? l : pl);
}

// ---------------------------------------------------------------------------
extern "C" void kernel_launch(void* const* d_in, const int* in_sizes, int n_in,
                              void* d_out, int out_size, void* d_ws, size_t ws_size,
                              hipStream_t stream) {
    const float* feat   = (const float*)d_in[0];   // [1024,512]
    const float* ipred  = (const float*)d_in[1];   // [1024]
    const int*   ilab   = (const int*)  d_in[2];   // [1024]
    const float* queue  = (const float*)d_in[3];   // [65536,512]
    const int*   qlab   = (const int*)  d_in[4];   // [65536]
    const float* protos = (const float*)d_in[5];   // [2,512]
    const float* pcc    = (const float*)d_in[6];   // [2,2]
    float* out = (float*)d_out;

    float* pos = (float*)d_ws;                     // [1024]
    float* neg = pos + B_TOT;                      // [1024]
    int*   prd = (int*)(neg + B_TOT);              // [1024]
    int*   cnt = prd + B_TOT;                      // [2]

    const size_t smem = (size_t)(QT * STRIDE + 2 * BT * STRIDE + QT + 2 * BT) * sizeof(float);
    (void)hipFuncSetAttribute(reinterpret_cast<const void*>(iwscl_gemm),
                              hipFuncAttributeMaxDynamicSharedMemorySize, (int)smem);

    iwscl_prep <<<1, B_TOT, 0, stream>>>(ipred, ilab, prd, pos, neg, cnt);
    iwscl_count<<<Q_TOT / 256, 256, 0, stream>>>(qlab, cnt);
    iwscl_gemm <<<Q_TOT / QT, NTHR, smem, stream>>>(feat, queue, qlab, prd, pos, neg);
    iwscl_final<<<1, B_TOT, 0, stream>>>(pos, neg, cnt, prd, ilab, feat, protos, pcc, out);
}